// MultiHeadAttention_49563922596691
// MI455X (gfx1250) — compile-verified
//
#include <hip/hip_runtime.h>
#include <hip/hip_bf16.h>

typedef __attribute__((ext_vector_type(2))) float v2f;
typedef __attribute__((ext_vector_type(8))) float v8f;

#define B_   8
#define T_   1024
#define C_   1024
#define H_   16
#define DK_  64
#define C3_  3072
#define LN_EPS 1e-5f

union V8 { v8f v; float f[8]; };

static __device__ __forceinline__ v8f wmma_f32(v2f a, v2f b, v8f c) {
  // V_WMMA_F32_16X16X4_F32 : D = A(16x4 f32) x B(4x16 f32) + C(16x16 f32)
  return __builtin_amdgcn_wmma_f32_16x16x4_f32(
      /*neg_a=*/false, a, /*neg_b=*/false, b,
      /*c_mod=*/(short)0, c, /*reuse_a=*/false, /*reuse_b=*/false);
}

// Async global->LDS copy of 16 bytes per lane (GLOBAL_LOAD_ASYNC_TO_LDS_B128,
// GV addressing, tracked by ASYNCcnt). LDS offset = low 32 bits of the generic
// pointer (flat->LDS mapping truncates to addr[31:0]).
static __device__ __forceinline__ void async_copy16(void* lds_dst,
                                                    const void* g_src) {
  unsigned ldsoff = (unsigned)(uintptr_t)lds_dst;
  asm volatile("global_load_async_to_lds_b128 %0, %1, off"
               :
               : "v"(ldsoff), "v"(g_src)
               : "memory");
}
static __device__ __forceinline__ void wait_async0() {
  asm volatile("s_wait_asynccnt 0x0" ::: "memory");
}

// ---------------------------------------------------------------------------
// Kernel 1: LayerNorm over rows of length 1024. One 256-thread block per row.
// ---------------------------------------------------------------------------
__global__ __launch_bounds__(256) void ln_kernel(
    const float* __restrict__ x, const float* __restrict__ g,
    const float* __restrict__ bta, float* __restrict__ h) {
  __shared__ float red[256];
  const int row = blockIdx.x;
  const int tid = threadIdx.x;
  const float4* xr = (const float4*)(x + (size_t)row * C_);
  float4 v = xr[tid];
  red[tid] = v.x + v.y + v.z + v.w;
  __syncthreads();
  for (int off = 128; off > 0; off >>= 1) {
    if (tid < off) red[tid] += red[tid + off];
    __syncthreads();
  }
  const float mu = red[0] * (1.0f / (float)C_);
  __syncthreads();
  const float dx = v.x - mu, dy = v.y - mu, dz = v.z - mu, dw = v.w - mu;
  red[tid] = dx * dx + dy * dy + dz * dz + dw * dw;
  __syncthreads();
  for (int off = 128; off > 0; off >>= 1) {
    if (tid < off) red[tid] += red[tid + off];
    __syncthreads();
  }
  const float rstd = rsqrtf(red[0] * (1.0f / (float)C_) + LN_EPS);
  const float4 gv = ((const float4*)g)[tid];
  const float4 bv = ((const float4*)bta)[tid];
  float4 o;
  o.x = dx * rstd * gv.x + bv.x;
  o.y = dy * rstd * gv.y + bv.y;
  o.z = dz * rstd * gv.z + bv.z;
  o.w = dw * rstd * gv.w + bv.w;
  ((float4*)(h + (size_t)row * C_))[tid] = o;
}

// ---------------------------------------------------------------------------
// Kernel 2/4: fp32 WMMA GEMM  C[M,N] = A[M,K] @ W[K,N] + bias (+ residual)
// 64x64 block tile, 8 waves; double-buffered async global->LDS staging so the
// chunk c+1 copy overlaps WMMA compute on chunk c.
// ---------------------------------------------------------------------------
__global__ __launch_bounds__(256) void gemm_kernel(
    const float* __restrict__ A, const float* __restrict__ W,
    const float* __restrict__ bias, const float* __restrict__ res,
    float* __restrict__ Cout, int M, int N, int K) {
  __shared__ float As[2][64 * 16];
  __shared__ float Bs[2][16 * 64];
  const int tid  = threadIdx.x;
  const int lane = tid & 31;
  const int wv   = tid >> 5;
  const int m0 = blockIdx.y * 64;
  const int n0 = blockIdx.x * 64;
  const int mi = wv & 3;        // 16-row band within 64
  const int nj = wv >> 2;       // 32-col band within 64
  const int kh   = (lane >> 4) * 2;   // K-half select (lanes 16-31 -> K+2)
  const int cola = lane & 15;
  const int mrow = mi * 16 + cola;
  // staging indices: one float4 per thread per tile
  const int ar = tid >> 2, ac = (tid & 3) * 4;    // A: 64 rows x 16 cols
  const int br = tid >> 4, bc = (tid & 15) * 4;   // B: 16 rows x 64 cols

  V8 acc0, acc1;
  for (int i = 0; i < 8; ++i) { acc0.f[i] = 0.f; acc1.f[i] = 0.f; }

  // prologue: async-stage chunk 0
  async_copy16(&As[0][ar * 16 + ac], &A[(size_t)(m0 + ar) * K + ac]);
  async_copy16(&Bs[0][br * 64 + bc], &W[(size_t)br * N + n0 + bc]);

  const int nchunks = K >> 4;
  for (int c = 0; c < nchunks; ++c) {
    wait_async0();
    __syncthreads();
    if (c + 1 < nchunks) {
      const int k0 = (c + 1) << 4;
      async_copy16(&As[(c + 1) & 1][ar * 16 + ac],
                   &A[(size_t)(m0 + ar) * K + k0 + ac]);
      async_copy16(&Bs[(c + 1) & 1][br * 64 + bc],
                   &W[(size_t)(k0 + br) * N + n0 + bc]);
    }
    const float* as = As[c & 1];
    const float* bs = Bs[c & 1];
#pragma unroll
    for (int kk = 0; kk < 16; kk += 4) {
      v2f a = *(const v2f*)&as[mrow * 16 + kk + kh];
      {
        const int col = nj * 32 + cola;
        v2f b;
        b.x = bs[(kk + kh) * 64 + col];
        b.y = bs[(kk + kh + 1) * 64 + col];
        acc0.v = wmma_f32(a, b, acc0.v);
      }
      {
        const int col = nj * 32 + 16 + cola;
        v2f b;
        b.x = bs[(kk + kh) * 64 + col];
        b.y = bs[(kk + kh + 1) * 64 + col];
        acc1.v = wmma_f32(a, b, acc1.v);
      }
    }
  }

  const int c0 = n0 + nj * 32 + cola;
  const int c1 = c0 + 16;
  const float b0 = bias[c0], b1 = bias[c1];
#pragma unroll
  for (int r = 0; r < 8; ++r) {
    const int row = m0 + mi * 16 + r + (lane >> 4) * 8;
    float v0 = acc0.f[r] + b0;
    float v1 = acc1.f[r] + b1;
    if (res) {
      // final FC: + residual, and `out` is never re-read -> non-temporal
      v0 += res[(size_t)row * N + c0];
      v1 += res[(size_t)row * N + c1];
      __builtin_nontemporal_store(v0, &Cout[(size_t)row * N + c0]);
      __builtin_nontemporal_store(v1, &Cout[(size_t)row * N + c1]);
    } else {
      Cout[(size_t)row * N + c0] = v0;   // qkv is re-read: keep cached
      Cout[(size_t)row * N + c1] = v1;
    }
  }
}

// ---------------------------------------------------------------------------
// Kernel 3: attention for one (b, h, 64-row q block).
// 304 KB LDS: S[64][1024] score stripe + Qs[64][64] + double-buffered
// KV[2][64][64]. scores (WMMA) -> softmax (wave32 shuffles) -> one-shot NT
// attn write -> P@V (WMMA). K/V chunks stream via async global->LDS copies
// overlapped with compute; V chunk 0 overlaps the softmax phase.
// ---------------------------------------------------------------------------
#define ATTN_SMEM_FLOATS (64 * 1024 + 64 * 64 + 2 * 64 * 64)
#define ATTN_SMEM_BYTES  (ATTN_SMEM_FLOATS * 4)

__global__ __launch_bounds__(256) void attn_kernel(
    const float* __restrict__ qkv, float* __restrict__ attn,
    float* __restrict__ o) {
  extern __shared__ float smem[];
  float* S  = smem;                  // 64 x 1024
  float* Qs = smem + 64 * 1024;      // 64 x 64
  float* KV0 = Qs + 64 * 64;         // 64 x 64 (double buffer 0)
  float* KV1 = KV0 + 64 * 64;        // 64 x 64 (double buffer 1)

  const int tid  = threadIdx.x;
  const int lane = tid & 31;
  const int wv   = tid >> 5;
  const int m0 = blockIdx.x * 64;
  const int hy = blockIdx.y;
  const int bz = blockIdx.z;
  const float scale = 0.125f;  // 1/sqrt(64)

  const int mi = wv & 3, nj = wv >> 2;
  const int kh = (lane >> 4) * 2;
  const int cola = lane & 15;
  const int mrow = mi * 16 + cola;
  const int ldr = tid >> 2, ldc = (tid & 3) * 16;  // 64x64 tile loader
  const int col0 = nj * 32 + cola, col1 = col0 + 16;

  // ---- async-stage Q block and K chunk 0 ----
  {
    const float* src = qkv + (size_t)(bz * T_ + m0 + ldr) * C3_ + hy * DK_ + ldc;
    float* dst = Qs + ldr * 64 + ldc;
#pragma unroll
    for (int i = 0; i < 4; ++i) async_copy16(dst + 4 * i, src + 4 * i);
  }
  {
    const float* src = qkv + (size_t)(bz * T_ + ldr) * C3_ + C_ + hy * DK_ + ldc;
    float* dst = KV0 + ldr * 64 + ldc;
#pragma unroll
    for (int i = 0; i < 4; ++i) async_copy16(dst + 4 * i, src + 4 * i);
  }

  // ---- S = scale * Q @ K^T, streamed into the LDS stripe ----
  for (int nb = 0; nb < 16; ++nb) {
    wait_async0();
    __syncthreads();
    if (nb + 1 < 16) {
      const int n0n = (nb + 1) * 64;
      const float* src =
          qkv + (size_t)(bz * T_ + n0n + ldr) * C3_ + C_ + hy * DK_ + ldc;
      float* dst = ((nb + 1) & 1 ? KV1 : KV0) + ldr * 64 + ldc;
#pragma unroll
      for (int i = 0; i < 4; ++i) async_copy16(dst + 4 * i, src + 4 * i);
    }
    const float* kv = (nb & 1) ? KV1 : KV0;
    V8 s0, s1;
    for (int i = 0; i < 8; ++i) { s0.f[i] = 0.f; s1.f[i] = 0.f; }
#pragma unroll
    for (int kd = 0; kd < 64; kd += 4) {
      v2f a  = *(const v2f*)&Qs[mrow * 64 + kd + kh];
      v2f b0 = *(const v2f*)&kv[col0 * 64 + kd + kh];  // K[n][d] = B[d][n]
      v2f b1 = *(const v2f*)&kv[col1 * 64 + kd + kh];
      s0.v = wmma_f32(a, b0, s0.v);
      s1.v = wmma_f32(a, b1, s1.v);
    }
    const int n0 = nb * 64;
#pragma unroll
    for (int r = 0; r < 8; ++r) {
      const int row = mi * 16 + r + (lane >> 4) * 8;
      S[row * 1024 + n0 + col0] = s0.f[r] * scale;
      S[row * 1024 + n0 + col1] = s1.f[r] * scale;
    }
  }
  __syncthreads();

  // ---- overlap: start V chunk 0 copy while softmax runs ----
  {
    const float* src =
        qkv + (size_t)(bz * T_ + ldr) * C3_ + 2 * C_ + hy * DK_ + ldc;
    float* dst = KV0 + ldr * 64 + ldc;
#pragma unroll
    for (int i = 0; i < 4; ++i) async_copy16(dst + 4 * i, src + 4 * i);
  }

  // ---- row softmax in LDS; write attn to HBM exactly once (non-temporal:
  //      512MB stream, never re-read; keep L2 for qkv) ----
  for (int rr = 0; rr < 8; ++rr) {
    const int row = wv * 8 + rr;
    float* srow = S + row * 1024;
    float mx = -3.402823466e38f;
    for (int j = lane; j < 1024; j += 32) mx = fmaxf(mx, srow[j]);
#pragma unroll
    for (int off = 16; off > 0; off >>= 1) mx = fmaxf(mx, __shfl_xor(mx, off));
    float sum = 0.f;
    for (int j = lane; j < 1024; j += 32) {
      const float e = __expf(srow[j] - mx);
      srow[j] = e;
      sum += e;
    }
#pragma unroll
    for (int off = 16; off > 0; off >>= 1) sum += __shfl_xor(sum, off);
    const float inv = 1.0f / sum;
    float* arow = attn + (size_t)((bz * H_ + hy) * T_ + m0 + row) * T_;
    for (int j = lane; j < 1024; j += 32) {
      const float p = srow[j] * inv;
      srow[j] = p;
      __builtin_nontemporal_store(p, &arow[j]);
    }
  }

  // ---- O = P @ V (V chunks double-buffered, async) ----
  V8 o0, o1;
  for (int i = 0; i < 8; ++i) { o0.f[i] = 0.f; o1.f[i] = 0.f; }
  for (int kb = 0; kb < 16; ++kb) {
    wait_async0();
    __syncthreads();
    if (kb + 1 < 16) {
      const int k0n = (kb + 1) * 64;
      const float* src =
          qkv + (size_t)(bz * T_ + k0n + ldr) * C3_ + 2 * C_ + hy * DK_ + ldc;
      float* dst = ((kb + 1) & 1 ? KV1 : KV0) + ldr * 64 + ldc;
#pragma unroll
      for (int i = 0; i < 4; ++i) async_copy16(dst + 4 * i, src + 4 * i);
    }
    const float* kv = (kb & 1) ? KV1 : KV0;
    const int k0 = kb * 64;
#pragma unroll
    for (int kk = 0; kk < 64; kk += 4) {
      v2f a = *(const v2f*)&S[mrow * 1024 + k0 + kk + kh];
      v2f b0, b1;
      b0.x = kv[(kk + kh) * 64 + col0];
      b0.y = kv[(kk + kh + 1) * 64 + col0];
      b1.x = kv[(kk + kh) * 64 + col1];
      b1.y = kv[(kk + kh + 1) * 64 + col1];
      o0.v = wmma_f32(a, b0, o0.v);
      o1.v = wmma_f32(a, b1, o1.v);
    }
  }
#pragma unroll
  for (int r = 0; r < 8; ++r) {
    const int row = m0 + mi * 16 + r + (lane >> 4) * 8;
    const int c0 = hy * DK_ + col0;
    o[(size_t)(bz * T_ + row) * C_ + c0]      = o0.f[r];  // re-read by FC GEMM
    o[(size_t)(bz * T_ + row) * C_ + c0 + 16] = o1.f[r];
  }
}

// ---------------------------------------------------------------------------
extern "C" void kernel_launch(void* const* d_in, const int* in_sizes, int n_in,
                              void* d_out, int out_size, void* d_ws,
                              size_t ws_size, hipStream_t stream) {
  const float* x     = (const float*)d_in[0];
  const float* ln_g  = (const float*)d_in[1];
  const float* ln_b  = (const float*)d_in[2];
  const float* W_qkv = (const float*)d_in[3];
  const float* b_qkv = (const float*)d_in[4];
  const float* W_fc  = (const float*)d_in[5];
  const float* b_fc  = (const float*)d_in[6];

  float* out  = (float*)d_out;                    // [8,1024,1024]
  float* attn = out + (size_t)B_ * T_ * C_;       // [8,16,1024,1024]

  float* ws  = (float*)d_ws;
  float* h   = ws;                                         // 8192x1024
  float* qkv = ws + (size_t)B_ * T_ * C_;                  // 8192x3072
  float* o   = qkv + (size_t)B_ * T_ * C3_;                // 8192x1024

  // 1. LayerNorm
  ln_kernel<<<B_ * T_, 256, 0, stream>>>(x, ln_g, ln_b, h);

  // 2. QKV projection: [8192,1024] @ [1024,3072] + b_qkv
  gemm_kernel<<<dim3(C3_ / 64, (B_ * T_) / 64), 256, 0, stream>>>(
      h, W_qkv, b_qkv, nullptr, qkv, B_ * T_, C3_, C_);

  // 3. Attention with 304KB-LDS resident score stripe
  (void)hipFuncSetAttribute(reinterpret_cast<const void*>(&attn_kernel),
                            hipFuncAttributeMaxDynamicSharedMemorySize,
                            ATTN_SMEM_BYTES);
  attn_kernel<<<dim3(T_ / 64, H_, B_), 256, ATTN_SMEM_BYTES, stream>>>(qkv,
                                                                       attn, o);

  // 4. FC + bias + residual: [8192,1024] @ [1024,1024] + b_fc + x
  gemm_kernel<<<dim3(C_ / 64, (B_ * T_) / 64), 256, 0, stream>>>(
      o, W_fc, b_fc, x, out, B_ * T_, C_, C_);
}